// MACFreeBlock_58042188038469
// MI455X (gfx1250) — compile-verified
//
#include <hip/hip_runtime.h>

// CDNA5 (gfx1250) fused soft-LBP block: proj -> bits -> fuse(1x1) -> BN -> +proj_res -> ReLU
// All three 1x1-conv GEMMs run on V_WMMA_F32_16X16X4_F32 (wave32 WMMA).

typedef float v2f __attribute__((ext_vector_type(2)));
typedef float v8f __attribute__((ext_vector_type(8)));

#define BATCH 16
#define CIN   32
#define HH    128
#define WW    128
#define PP    4
#define NNB   8
#define COUT  64
#define BN_EPS_C 1e-5f
// ALPHA * log2(e): sigmoid(a*d) = rcp(1 + exp2(-a*log2e*d))
#define ALPHA_LOG2E 14.4269504089f

#define HB      4      // image rows per workgroup slab
#define NWAVES  8
#define THREADS 256

// ---- LDS layout (float offsets) ----
#define OFF_WPAT  0                       // 16 x 33 (w_pattern zero-padded to 16 rows)
#define OFF_FUSE  (OFF_WPAT + 16*33)      // 64 x 33
#define OFF_PROJW (OFF_FUSE + 64*33)      // 64 x 33
#define OFF_BN    (OFF_PROJW + 64*33)     // inv[64], add[64]
#define PROJ_RS   132                     // padded row stride (cols)
#define PROJ_PS   ((HB+2)*PROJ_RS)        // per-pattern plane (6 halo rows)
#define OFF_PROJ  (OFF_BN + 128)          // 4 * 6 * 132
#define BITS_PS   528                     // HB*WW=512 + 16 pad (2*528 % 64 == 32 -> no bank clash)
#define OFF_BITS  (OFF_PROJ + PP*PROJ_PS) // 32 * 528
#define LDS_FLOATS (OFF_BITS + CIN*BITS_PS)   // = 24944 floats = 99776 B

// packed (dy+1),(dx+1) 2-bit tables for the 8-neighbor ring
// OFFSETS = (-1,-1)(-1,0)(-1,1)(0,1)(1,1)(1,0)(1,-1)(0,-1)
#define DYPACK 0x6A40u
#define DXPACK 0x06A4u

__global__ __launch_bounds__(THREADS)
void lbp_fused_kernel(const float* __restrict__ x,
                      const float* __restrict__ wpat,
                      const float* __restrict__ fusew,
                      const float* __restrict__ gammaw,
                      const float* __restrict__ betaw,
                      const float* __restrict__ meanw,
                      const float* __restrict__ varw,
                      const float* __restrict__ projw,
                      const float* __restrict__ res_scale,
                      float* __restrict__ out)
{
    extern __shared__ float lds[];
    const int tid  = threadIdx.x;
    const int lane = tid & 31;
    const int wave = tid >> 5;
    const int hl   = lane >> 4;   // half-wave select (K split)
    const int ln   = lane & 15;   // M/N index within tile

    const int blk  = blockIdx.x;
    const int bimg = blk >> 5;    // 32 h-slabs per image
    const int hblk = blk & 31;
    const int h0   = hblk * HB;

    // ---------------- phase 0: stage weights + folded BN into LDS ----------------
    for (int i = tid; i < 16*CIN; i += THREADS) {
        int m = i >> 5, k = i & 31;
        lds[OFF_WPAT + m*33 + k] = (m < PP) ? wpat[m*CIN + k] : 0.0f;
    }
    for (int i = tid; i < COUT*CIN; i += THREADS) {
        int o = i >> 5, k = i & 31;
        lds[OFF_FUSE  + o*33 + k] = fusew[i];
        lds[OFF_PROJW + o*33 + k] = projw[i];
    }
    if (tid < COUT) {
        float inv = gammaw[tid] * rsqrtf(varw[tid] + BN_EPS_C);
        lds[OFF_BN + tid]      = inv;
        lds[OFF_BN + 64 + tid] = betaw[tid] - meanw[tid] * inv;
    }
    __syncthreads();

    const float* xb = x + (size_t)bimg * (CIN*HH*WW);

    // ---------------- phase 1: proj (slab + circular halo) via WMMA ----------------
    // D[p][pix] = sum_c wpat[p][c] * x[c][pix] ; A = padded w_pattern (16x32), B = x (32x16 pixels)
    for (int t = wave; t < (HB+2)*8; t += NWAVES) {
        int prow = t >> 3;                       // 0..5 (halo row)
        int w0   = (t & 7) << 4;
        int hg   = (h0 - 1 + prow) & (HH - 1);   // circular in H
        v8f acc = {};
        #pragma unroll
        for (int ks = 0; ks < CIN; ks += 4) {
            int kb = ks + 2*hl;
            v2f a, bm;
            a[0] = lds[OFF_WPAT + ln*33 + kb];
            a[1] = lds[OFF_WPAT + ln*33 + kb + 1];
            const float* xp = xb + ((size_t)kb*HH + hg)*WW + w0 + ln;
            bm[0] = xp[0];
            bm[1] = xp[HH*WW];
            acc = __builtin_amdgcn_wmma_f32_16x16x4_f32(false, a, false, bm,
                                                        (short)0, acc, false, false);
        }
        if (hl == 0) {   // only D rows M=0..3 are real patterns
            #pragma unroll
            for (int r = 0; r < PP; ++r)
                lds[OFF_PROJ + r*PROJ_PS + prow*PROJ_RS + w0 + ln] = acc[r];
        }
    }
    __syncthreads();

    // ---------------- phase 2a: soft LBP bits into LDS ----------------
    // bits[k=p*8+n][pix] = sigmoid(ALPHA * (proj[h-dy][w-dx] - proj[h][w]))  (roll semantics)
    // sigmoid(a*d) = rcp(1 + exp2(a*log2e*(c - nb)))  -- native v_exp_f32 + v_rcp_f32
    for (int i = tid; i < CIN*HB*WW; i += THREADS) {
        int k   = i >> 9;                 // /512
        int pix = i & 511;
        int row = pix >> 7, col = pix & 127;
        int p = k >> 3, nf = k & 7;
        int dy = (int)((DYPACK >> (2*nf)) & 3u) - 1;
        int dx = (int)((DXPACK >> (2*nf)) & 3u) - 1;
        float c  = lds[OFF_PROJ + p*PROJ_PS + (row+1)*PROJ_RS + col];
        float nb = lds[OFF_PROJ + p*PROJ_PS + (row+1-dy)*PROJ_RS + ((col - dx) & (WW-1))];
        float t  = __builtin_amdgcn_exp2f(ALPHA_LOG2E * (c - nb));
        lds[OFF_BITS + k*BITS_PS + pix] = __builtin_amdgcn_rcpf(1.0f + t);
    }
    __syncthreads();

    const float srs = res_scale[0];

    // ---------------- phase 2b: fused + identity GEMMs, BN, residual, ReLU ----------------
    for (int t = wave; t < HB*8; t += NWAVES) {
        int row  = t >> 3;
        int w0   = (t & 7) << 4;
        int hg   = h0 + row;
        int pixb = (row << 7) + w0;

        v8f accF[4] = {};   // fused  (fuse_w x bits)
        v8f accI[4] = {};   // ident  (proj_w x x)
        #pragma unroll
        for (int ks = 0; ks < CIN; ks += 4) {
            int kb = ks + 2*hl;
            v2f bb, bx;
            bb[0] = lds[OFF_BITS + kb*BITS_PS + pixb + ln];
            bb[1] = lds[OFF_BITS + (kb+1)*BITS_PS + pixb + ln];
            const float* xp = xb + ((size_t)kb*HH + hg)*WW + w0 + ln;
            bx[0] = xp[0];
            bx[1] = xp[HH*WW];
            #pragma unroll
            for (int mt = 0; mt < 4; ++mt) {
                int ro = (mt*16 + ln)*33 + kb;
                v2f aF, aI;
                aF[0] = lds[OFF_FUSE  + ro]; aF[1] = lds[OFF_FUSE  + ro + 1];
                aI[0] = lds[OFF_PROJW + ro]; aI[1] = lds[OFF_PROJW + ro + 1];
                accF[mt] = __builtin_amdgcn_wmma_f32_16x16x4_f32(false, aF, false, bb,
                                                                 (short)0, accF[mt], false, false);
                accI[mt] = __builtin_amdgcn_wmma_f32_16x16x4_f32(false, aI, false, bx,
                                                                 (short)0, accI[mt], false, false);
            }
        }

        // epilogue: lane holds D[o = mt*16 + r + 8*hl][pixel = w0+ln] -> coalesced stores
        float* op = out + (((size_t)bimg*COUT)*HH + hg)*WW + w0 + ln;
        #pragma unroll
        for (int mt = 0; mt < 4; ++mt) {
            #pragma unroll
            for (int r = 0; r < 8; ++r) {
                int o = mt*16 + r + 8*hl;
                float v = (accF[mt][r]*lds[OFF_BN + o] + lds[OFF_BN + 64 + o])*srs + accI[mt][r];
                op[(size_t)o*HH*WW] = v > 0.0f ? v : 0.0f;
            }
        }
    }
}

extern "C" void kernel_launch(void* const* d_in, const int* in_sizes, int n_in,
                              void* d_out, int out_size, void* d_ws, size_t ws_size,
                              hipStream_t stream) {
    (void)in_sizes; (void)n_in; (void)d_ws; (void)ws_size; (void)out_size;
    const float* x      = (const float*)d_in[0];
    const float* wpat   = (const float*)d_in[1];
    const float* fusew  = (const float*)d_in[2];
    const float* gammaw = (const float*)d_in[3];
    const float* betaw  = (const float*)d_in[4];
    const float* meanw  = (const float*)d_in[5];
    const float* varw   = (const float*)d_in[6];
    const float* projw  = (const float*)d_in[7];
    const float* rscale = (const float*)d_in[8];
    float* outp = (float*)d_out;

    dim3 grid(BATCH * (HH / HB));        // 512 workgroups
    dim3 block(THREADS);                 // 8 waves (wave32)
    size_t ldsBytes = (size_t)LDS_FLOATS * sizeof(float);   // ~97.4 KB < 320 KB/WGP
    lbp_fused_kernel<<<grid, block, ldsBytes, stream>>>(
        x, wpat, fusew, gammaw, betaw, meanw, varw, projw, rscale, outp);
}